// LDR_6828998000734
// MI455X (gfx1250) — compile-verified
//
#include <hip/hip_runtime.h>
#include <hip/hip_bf16.h>

// LDR toeplitz-corner forward on gfx1250:
// all length-4096 FFTs are done as 64x64 Cooley-Tukey, i.e. two complex
// 64x64x64 matmuls (8 real matmuls) executed with V_WMMA_F32_16X16X4_F32,
// DFT-matrix fragments generated on the fly via __sincosf with exact
// integer angle reduction. Frequency-domain layout is the permuted (k1,k2)
// matrix layout; it is self-consistent because every frequency-domain op in
// the pipeline is elementwise and ifft_mat exactly inverts fft_mat.

typedef __attribute__((ext_vector_type(2))) float v2f;
typedef __attribute__((ext_vector_type(8))) float v8f;

#define DIM   64
#define NFFT  4096
#define TPB   512          // 16 waves of 32; each wave owns one 16x16 D tile
#define PI_F  3.14159265358979323846f

#define CIN   3
#define COUT  3
#define RANK  4
#define BATCH 64

// ---------------------------------------------------------------------------
// store a complex 16x16 WMMA accumulator tile to LDS (C/D layout: VGPR v holds
// M=v for lanes 0-15 and M=v+8 for lanes 16-31, N = lane&15)
__device__ __forceinline__ void wmma_store_tile(v8f cr, v8f ci,
                                                float* dR, float* dI,
                                                int m0, int n0, int lane) {
  const int mo = m0 + ((lane >> 4) << 3);
  const int nc = n0 + (lane & 15);
#pragma unroll
  for (int v = 0; v < 8; ++v) {
    dR[(mo + v) * DIM + nc] = cr[v];
    dI[(mo + v) * DIM + nc] = ci[v];
  }
}

// dst = F(sgn) * src, F[a][b] = exp(i*sgn*2*pi*a*b/64), complex 64x64x64.
// src/dst are separate LDS planes. Trailing barrier leaves dst visible.
__device__ __forceinline__ void dft_left(float* dR, float* dI,
                                         const float* sR, const float* sI,
                                         float sgn) {
  const int tid  = threadIdx.x;
  const int wv   = tid >> 5, lane = tid & 31;
  const int m0   = (wv >> 2) * 16, n0 = (wv & 3) * 16;
  const int kb   = (lane >> 4) << 1;          // A/B fragment K sub-offset
  const int m    = m0 + (lane & 15);          // A-fragment row
  const int n    = n0 + (lane & 15);          // B-fragment col
  const float w  = sgn * (2.0f * PI_F / 64.0f);
  v8f accRR = {}, accII = {}, accRI = {}, accIR = {};
  for (int s = 0; s < 16; ++s) {
    const int k = s * 4 + kb;
    float s0, c0, s1, c1;
    __sincosf(w * (float)((m * k) & 63), &s0, &c0);
    __sincosf(w * (float)((m * (k + 1)) & 63), &s1, &c1);
    v2f aFr = {c0, c1};
    v2f aFi = {s0, s1};
    v2f bSr = {sR[k * DIM + n], sR[(k + 1) * DIM + n]};
    v2f bSi = {sI[k * DIM + n], sI[(k + 1) * DIM + n]};
    accRR = __builtin_amdgcn_wmma_f32_16x16x4_f32(false, aFr, false, bSr, (short)0, accRR, false, false);
    accII = __builtin_amdgcn_wmma_f32_16x16x4_f32(false, aFi, false, bSi, (short)0, accII, false, false);
    accRI = __builtin_amdgcn_wmma_f32_16x16x4_f32(false, aFr, false, bSi, (short)0, accRI, false, false);
    accIR = __builtin_amdgcn_wmma_f32_16x16x4_f32(false, aFi, false, bSr, (short)0, accIR, false, false);
  }
  wmma_store_tile(accRR - accII, accRI + accIR, dR, dI, m0, n0, lane);
  __syncthreads();
}

// dst = src * F(sgn)  (F symmetric), complex 64x64x64.
__device__ __forceinline__ void dft_right(float* dR, float* dI,
                                          const float* sR, const float* sI,
                                          float sgn) {
  const int tid  = threadIdx.x;
  const int wv   = tid >> 5, lane = tid & 31;
  const int m0   = (wv >> 2) * 16, n0 = (wv & 3) * 16;
  const int kb   = (lane >> 4) << 1;
  const int m    = m0 + (lane & 15);
  const int n    = n0 + (lane & 15);
  const float w  = sgn * (2.0f * PI_F / 64.0f);
  v8f accRR = {}, accII = {}, accRI = {}, accIR = {};
  for (int s = 0; s < 16; ++s) {
    const int k = s * 4 + kb;
    v2f aSr = {sR[m * DIM + k], sR[m * DIM + k + 1]};
    v2f aSi = {sI[m * DIM + k], sI[m * DIM + k + 1]};
    float s0, c0, s1, c1;
    __sincosf(w * (float)((k * n) & 63), &s0, &c0);
    __sincosf(w * (float)(((k + 1) * n) & 63), &s1, &c1);
    v2f bFr = {c0, c1};
    v2f bFi = {s0, s1};
    accRR = __builtin_amdgcn_wmma_f32_16x16x4_f32(false, aSr, false, bFr, (short)0, accRR, false, false);
    accII = __builtin_amdgcn_wmma_f32_16x16x4_f32(false, aSi, false, bFi, (short)0, accII, false, false);
    accRI = __builtin_amdgcn_wmma_f32_16x16x4_f32(false, aSr, false, bFi, (short)0, accRI, false, false);
    accIR = __builtin_amdgcn_wmma_f32_16x16x4_f32(false, aSi, false, bFr, (short)0, accIR, false, false);
  }
  wmma_store_tile(accRR - accII, accRI + accIR, dR, dI, m0, n0, lane);
  __syncthreads();
}

// elementwise: X[a][b] *= scale * exp(i*sgn*2*pi*a*b/4096)  (in place)
__device__ __forceinline__ void twiddle4096(float* r, float* im, float sgn, float scale) {
  const int tid = threadIdx.x;
#pragma unroll
  for (int q = 0; q < 8; ++q) {
    const int e = tid * 8 + q;
    const int a = e >> 6, b = e & 63;
    float sn, cs;
    __sincosf(sgn * (2.0f * PI_F / 4096.0f) * (float)(a * b), &sn, &cs);
    const float xr = r[e], xi = im[e];
    r[e]  = (xr * cs - xi * sn) * scale;
    im[e] = (xr * sn + xi * cs) * scale;
  }
  __syncthreads();
}

// forward fft (jnp convention, exp(-i...)): A <- FFT(A), using B as temp
__device__ __forceinline__ void fft_mat(float* Ar, float* Ai, float* Br, float* Bi) {
  dft_left(Br, Bi, Ar, Ai, -1.0f);
  twiddle4096(Br, Bi, -1.0f, 1.0f);
  dft_right(Ar, Ai, Br, Bi, -1.0f);
}

// inverse fft with 1/N: A <- IFFT(A), using B as temp
__device__ __forceinline__ void ifft_mat(float* Ar, float* Ai, float* Br, float* Bi) {
  dft_right(Br, Bi, Ar, Ai, 1.0f);
  twiddle4096(Br, Bi, 1.0f, 1.0f / (float)NFFT);
  dft_left(Ar, Ai, Br, Bi, 1.0f);
}

// ---------------------------------------------------------------------------
// K1: fwc[i,j,r] = conj(fft(conj(eta) * H[i,j,r])) ; fG[i,j,r] = fft(G[i,j,r])
__global__ __launch_bounds__(TPB) void ldr_prep_weights(const float* __restrict__ G,
                                                        const float* __restrict__ H,
                                                        float2* __restrict__ fG,
                                                        float2* __restrict__ fwc) {
  __shared__ float Ar[NFFT], Ai[NFFT], Br[NFFT], Bi[NFFT];
  const int idx = blockIdx.x;                 // (i*COUT + j)*RANK + r
  const int tid = threadIdx.x;
  const float* h = H + (size_t)idx * NFFT;
  const float* g = G + (size_t)idx * NFFT;

  // pass 1: conj(eta)*H, fft, conjugate-store
#pragma unroll
  for (int q = 0; q < 8; ++q) {
    const int e = tid * 8 + q;
    float sn, cs;
    __sincosf(PI_F * (float)e / (float)NFFT, &sn, &cs);
    const float hv = h[e];
    Ar[e] = hv * cs;          // Re(conj(eta)*H)
    Ai[e] = -hv * sn;         // Im(conj(eta)*H)
  }
  __syncthreads();
  fft_mat(Ar, Ai, Br, Bi);
  float2* ow = fwc + (size_t)idx * NFFT;
#pragma unroll
  for (int q = 0; q < 8; ++q) {
    const int e = tid * 8 + q;
    ow[e] = make_float2(Ar[e], -Ai[e]);       // conjugate
  }
  __syncthreads();

  // pass 2: fG = fft(G)
#pragma unroll
  for (int q = 0; q < 8; ++q) {
    const int e = tid * 8 + q;
    Ar[e] = g[e];
    Ai[e] = 0.0f;
  }
  __syncthreads();
  fft_mat(Ar, Ai, Br, Bi);
  float2* og = fG + (size_t)idx * NFFT;
#pragma unroll
  for (int q = 0; q < 8; ++q) {
    const int e = tid * 8 + q;
    og[e] = make_float2(Ar[e], Ai[e]);
  }
}

// K2: fz[i,b] = fft(x[i,b] * conj(eta))
__global__ __launch_bounds__(TPB) void ldr_fft_x(const float* __restrict__ x,
                                                 float2* __restrict__ fz) {
  __shared__ float Ar[NFFT], Ai[NFFT], Br[NFFT], Bi[NFFT];
  const int idx = blockIdx.x;                 // i*BATCH + b
  const int tid = threadIdx.x;
  const float* xp = x + (size_t)idx * NFFT;
#pragma unroll
  for (int q = 0; q < 8; ++q) {
    const int e = tid * 8 + q;
    float sn, cs;
    __sincosf(PI_F * (float)e / (float)NFFT, &sn, &cs);
    const float xv = xp[e];
    Ar[e] = xv * cs;
    Ai[e] = -xv * sn;
  }
  __syncthreads();
  fft_mat(Ar, Ai, Br, Bi);
  float2* oz = fz + (size_t)idx * NFFT;
#pragma unroll
  for (int q = 0; q < 8; ++q) {
    const int e = tid * 8 + q;
    oz[e] = make_float2(Ar[e], Ai[e]);
  }
}

// K3: per (j,b): yf = sum_{i,r} fG[i,j,r] . fft(eta . ifft(fz[i,b] . fwc[i,j,r]))
//     out[j,b] = real(ifft(yf)) + bias[j]
__global__ __launch_bounds__(TPB) void ldr_apply(const float2* __restrict__ fz,
                                                 const float2* __restrict__ fwc,
                                                 const float2* __restrict__ fG,
                                                 const float* __restrict__ bias,
                                                 float* __restrict__ out) {
  __shared__ float Ar[NFFT], Ai[NFFT], Br[NFFT], Bi[NFFT];
  const int j   = blockIdx.x / BATCH;
  const int b   = blockIdx.x % BATCH;
  const int tid = threadIdx.x;

  float accR[8], accI[8];
#pragma unroll
  for (int q = 0; q < 8; ++q) { accR[q] = 0.0f; accI[q] = 0.0f; }

  for (int i = 0; i < CIN; ++i) {
    const float2* fzp = fz + (size_t)(i * BATCH + b) * NFFT;
    for (int r = 0; r < RANK; ++r) {
      const size_t wofs = (size_t)((i * COUT + j) * RANK + r) * NFFT;
      const float2* fw = fwc + wofs;
      // A <- fz . fwc (freq layout, elementwise)
#pragma unroll
      for (int q = 0; q < 8; ++q) {
        const int e = tid * 8 + q;
        const float2 zv = fzp[e];
        const float2 wv = fw[e];
        Ar[e] = zv.x * wv.x - zv.y * wv.y;
        Ai[e] = zv.x * wv.y + zv.y * wv.x;
      }
      __syncthreads();
      ifft_mat(Ar, Ai, Br, Bi);                 // t_pre (time layout)
      // A <- eta . A  (eta[n] = exp(i*pi*n/N), n == flattened index)
#pragma unroll
      for (int q = 0; q < 8; ++q) {
        const int e = tid * 8 + q;
        float sn, cs;
        __sincosf(PI_F * (float)e / (float)NFFT, &sn, &cs);
        const float xr = Ar[e], xi = Ai[e];
        Ar[e] = xr * cs - xi * sn;
        Ai[e] = xr * sn + xi * cs;
      }
      __syncthreads();
      fft_mat(Ar, Ai, Br, Bi);                  // ft (freq layout)
      // acc += fG . ft   (registers, per-thread-owned elements)
      const float2* fgp = fG + wofs;
#pragma unroll
      for (int q = 0; q < 8; ++q) {
        const int e = tid * 8 + q;
        const float2 gv = fgp[e];
        const float xr = Ar[e], xi = Ai[e];
        accR[q] += gv.x * xr - gv.y * xi;
        accI[q] += gv.x * xi + gv.y * xr;
      }
      // no barrier needed: each thread only re-writes its own elements next
    }
  }

  // final: y = ifft(acc); out = y.real + bias
#pragma unroll
  for (int q = 0; q < 8; ++q) {
    const int e = tid * 8 + q;
    Ar[e] = accR[q];
    Ai[e] = accI[q];
  }
  __syncthreads();
  ifft_mat(Ar, Ai, Br, Bi);
  const float* bp = bias + (size_t)j * NFFT;
  float* op = out + (size_t)(j * BATCH + b) * NFFT;
#pragma unroll
  for (int q = 0; q < 8; ++q) {
    const int e = tid * 8 + q;
    op[e] = Ar[e] + bp[e];
  }
}

// ---------------------------------------------------------------------------
extern "C" void kernel_launch(void* const* d_in, const int* in_sizes, int n_in,
                              void* d_out, int out_size, void* d_ws, size_t ws_size,
                              hipStream_t stream) {
  const float* x    = (const float*)d_in[0];   // (CIN, BATCH, N)
  const float* G    = (const float*)d_in[1];   // (CIN, COUT, RANK, N)
  const float* H    = (const float*)d_in[2];   // (CIN, COUT, RANK, N)
  const float* bias = (const float*)d_in[3];   // (COUT, 1, N)
  float* out = (float*)d_out;                  // (COUT, BATCH, N)

  // workspace layout (complex64 spectra, permuted (k1,k2) layout):
  //   fwc: CIN*COUT*RANK*N  (1.18 MB)
  //   fG : CIN*COUT*RANK*N  (1.18 MB)
  //   fz : CIN*BATCH*N      (6.29 MB)   -> total ~8.65 MB
  float2* fwc = (float2*)d_ws;
  float2* fG  = fwc + (size_t)CIN * COUT * RANK * NFFT;
  float2* fz  = fG  + (size_t)CIN * COUT * RANK * NFFT;

  ldr_prep_weights<<<CIN * COUT * RANK, TPB, 0, stream>>>(G, H, fG, fwc);
  ldr_fft_x<<<CIN * BATCH, TPB, 0, stream>>>(x, fz);
  ldr_apply<<<COUT * BATCH, TPB, 0, stream>>>(fz, fwc, fG, bias, out);
}